// SkeletonClassifier_33964601377213
// MI455X (gfx1250) — compile-verified
//
#include <hip/hip_runtime.h>
#include <hip/hip_bf16.h>

typedef __attribute__((ext_vector_type(2))) float v2f;
typedef __attribute__((ext_vector_type(8))) float v8f;

// ---------------------------------------------------------------------------
// degree / dinv
// ---------------------------------------------------------------------------
__global__ void gcn_init_deg(float* __restrict__ deg, int n) {
    int i = blockIdx.x * blockDim.x + threadIdx.x;
    if (i < n) deg[i] = 1.0f;              // self loop
}

__global__ void gcn_count_deg(const int* __restrict__ dst, float* __restrict__ deg, int e) {
    int i = blockIdx.x * blockDim.x + threadIdx.x;
    if (i < e) atomicAdd(&deg[dst[i]], 1.0f);
}

__global__ void gcn_finish_dinv(float* __restrict__ deg, int n) {
    int i = blockIdx.x * blockDim.x + threadIdx.x;
    if (i < n) deg[i] = rsqrtf(fmaxf(deg[i], 1.0f));
}

// ---------------------------------------------------------------------------
// C[M x 256] = act(A[M x K]) @ B[K x 256]   (exact f32 WMMA 16x16x4)
// One wave computes a 16x64 strip: A-fragment reused across 4 N-subtiles.
// Block = 256 threads = 8 waves: waves 0-3 -> n-subtiles of m-tile 0,
// waves 4-7 -> m-tile 1. Grid.x = M/32 (M % 32 == 0 here: 100000).
// ---------------------------------------------------------------------------
__global__ __launch_bounds__(256) void gcn_gemm_wmma_f32(
    const float* __restrict__ A, const float* __restrict__ B,
    float* __restrict__ C, int M, int K, int reluA)
{
    const int N    = 256;
    const int lane = threadIdx.x & 31;
    const int wave = threadIdx.x >> 5;
    const int nw   = wave & 3;                 // which 64-col strip
    const int mw   = wave >> 2;                // which 16-row tile in block
    const int m0   = blockIdx.x * 32 + mw * 16;
    if (m0 >= M) return;                       // wave-uniform guard

    const int hi = lane >> 4;                  // 0/1 half-wave
    const int ml = lane & 15;
    const int n0 = nw * 64;

    const float* __restrict__ arow = A + (size_t)(m0 + ml) * K;

    v8f acc0 = {}, acc1 = {}, acc2 = {}, acc3 = {};

    for (int k = 0; k < K; k += 4) {
        const int ka = k + 2 * hi;
        float a0 = arow[ka];
        float a1 = arow[ka + 1];
        if (reluA) { a0 = fmaxf(a0, 0.0f); a1 = fmaxf(a1, 0.0f); }
        v2f av = {a0, a1};

        const float* __restrict__ bp0 = B + (size_t)ka * N + n0 + ml;
        const float* __restrict__ bp1 = bp0 + N;

        v2f bv;
        bv[0] = bp0[0];  bv[1] = bp1[0];
        acc0 = __builtin_amdgcn_wmma_f32_16x16x4_f32(false, av, false, bv,
                                                     (short)0, acc0, false, false);
        bv[0] = bp0[16]; bv[1] = bp1[16];
        acc1 = __builtin_amdgcn_wmma_f32_16x16x4_f32(false, av, false, bv,
                                                     (short)0, acc1, false, false);
        bv[0] = bp0[32]; bv[1] = bp1[32];
        acc2 = __builtin_amdgcn_wmma_f32_16x16x4_f32(false, av, false, bv,
                                                     (short)0, acc2, false, false);
        bv[0] = bp0[48]; bv[1] = bp1[48];
        acc3 = __builtin_amdgcn_wmma_f32_16x16x4_f32(false, av, false, bv,
                                                     (short)0, acc3, false, false);
    }

    // D layout: VGPR r -> row (r + 8*hi), col = ml (plus n-subtile base)
#pragma unroll
    for (int r = 0; r < 8; ++r) {
        float* __restrict__ cp = C + (size_t)(m0 + r + 8 * hi) * N + n0 + ml;
        cp[0]  = acc0[r];
        cp[16] = acc1[r];
        cp[32] = acc2[r];
        cp[48] = acc3[r];
    }
}

// ---------------------------------------------------------------------------
// agg[i][c] = h[i][c] * dinv[i]^2 + bias[c]   (self-loop message + bias)
// ---------------------------------------------------------------------------
__global__ void gcn_init_agg(const float* __restrict__ h, const float* __restrict__ dinv,
                             const float* __restrict__ bias, float* __restrict__ agg,
                             int total)
{
    int idx = blockIdx.x * blockDim.x + threadIdx.x;
    if (idx >= total) return;
    const int node = idx >> 8;
    const int c    = idx & 255;
    const float di = dinv[node];
    agg[idx] = h[idx] * (di * di) + bias[c];
}

// ---------------------------------------------------------------------------
// per-edge scatter: agg[dst] += h[src] * dinv[src]*dinv[dst]
// one wave per edge, float4 gathers, f32 atomics (L2-resident)
// ---------------------------------------------------------------------------
__global__ __launch_bounds__(256) void gcn_scatter_edges(
    const float* __restrict__ h, float* __restrict__ agg,
    const int* __restrict__ src, const int* __restrict__ dst,
    const float* __restrict__ dinv, int e)
{
    const int w    = (int)((blockIdx.x * (size_t)blockDim.x + threadIdx.x) >> 5);
    const int lane = threadIdx.x & 31;
    if (w >= e) return;
    const int s = src[w];
    const int d = dst[w];
    const float norm = dinv[s] * dinv[d];
    const float4* __restrict__ hp = (const float4*)(h + (size_t)s * 256);
    float* __restrict__ ap = agg + (size_t)d * 256;
#pragma unroll
    for (int i = 0; i < 2; ++i) {
        const int q = lane + 32 * i;          // 0..63 float4 slots
        float4 v = hp[q];
        const int c = 4 * q;
        atomicAdd(ap + c + 0, v.x * norm);
        atomicAdd(ap + c + 1, v.y * norm);
        atomicAdd(ap + c + 2, v.z * norm);
        atomicAdd(ap + c + 3, v.w * norm);
    }
}

// ---------------------------------------------------------------------------
// pooling: sums[b] += relu(h[i]); cnt[b] += 1
// ---------------------------------------------------------------------------
__global__ void gcn_zero(float* __restrict__ p, int n) {
    int i = blockIdx.x * blockDim.x + threadIdx.x;
    if (i < n) p[i] = 0.0f;
}

__global__ __launch_bounds__(256) void gcn_pool(
    const float* __restrict__ h, const int* __restrict__ batch,
    float* __restrict__ sums, float* __restrict__ cnt, int n)
{
    const int w    = (int)((blockIdx.x * (size_t)blockDim.x + threadIdx.x) >> 5);
    const int lane = threadIdx.x & 31;
    if (w >= n) return;
    const int b = batch[w];
    const float4* __restrict__ hp = (const float4*)(h + (size_t)w * 256);
    float* __restrict__ sp = sums + (size_t)b * 256;
#pragma unroll
    for (int i = 0; i < 2; ++i) {
        const int q = lane + 32 * i;
        float4 v = hp[q];
        const int c = 4 * q;
        atomicAdd(sp + c + 0, fmaxf(v.x, 0.0f));
        atomicAdd(sp + c + 1, fmaxf(v.y, 0.0f));
        atomicAdd(sp + c + 2, fmaxf(v.z, 0.0f));
        atomicAdd(sp + c + 3, fmaxf(v.w, 0.0f));
    }
    if (lane == 0) atomicAdd(cnt + b, 1.0f);
}

__global__ void gcn_pool_div(const float* __restrict__ sums, const float* __restrict__ cnt,
                             float* __restrict__ g, int total)
{
    int idx = blockIdx.x * blockDim.x + threadIdx.x;
    if (idx >= total) return;
    g[idx] = sums[idx] / fmaxf(cnt[idx >> 8], 1.0f);
}

// ---------------------------------------------------------------------------
// classifier + log_softmax: one wave per graph, 2 of 64 logits per lane
// ---------------------------------------------------------------------------
__global__ __launch_bounds__(256) void gcn_classifier(
    const float* __restrict__ g, const float* __restrict__ Wc,
    const float* __restrict__ bc, float* __restrict__ out, int G)
{
    const int w    = (int)((blockIdx.x * (size_t)blockDim.x + threadIdx.x) >> 5);
    const int lane = threadIdx.x & 31;
    if (w >= G) return;
    const float* __restrict__ gr = g + (size_t)w * 256;
    float acc0 = bc[lane];
    float acc1 = bc[lane + 32];
    for (int k = 0; k < 256; ++k) {
        const float gv = gr[k];
        acc0 += gv * Wc[(size_t)k * 64 + lane];
        acc1 += gv * Wc[(size_t)k * 64 + lane + 32];
    }
    // log-softmax across 64 logits (2 per lane), wave32 shuffle reduction
    float m = fmaxf(acc0, acc1);
#pragma unroll
    for (int o = 16; o > 0; o >>= 1) m = fmaxf(m, __shfl_xor(m, o, 32));
    float s = expf(acc0 - m) + expf(acc1 - m);
#pragma unroll
    for (int o = 16; o > 0; o >>= 1) s += __shfl_xor(s, o, 32);
    const float lse = m + logf(s);
    out[(size_t)w * 64 + lane]      = acc0 - lse;
    out[(size_t)w * 64 + lane + 32] = acc1 - lse;
}

// ---------------------------------------------------------------------------
// launch
// ---------------------------------------------------------------------------
extern "C" void kernel_launch(void* const* d_in, const int* in_sizes, int n_in,
                              void* d_out, int out_size, void* d_ws, size_t ws_size,
                              hipStream_t stream)
{
    const float* x     = (const float*)d_in[0];   // [N,128]
    const int*   edges = (const int*)  d_in[1];   // [2,E]
    const int*   batch = (const int*)  d_in[2];   // [N]
    const float* W1    = (const float*)d_in[3];   // [128,256]
    const float* b1    = (const float*)d_in[4];   // [256]
    const float* W2    = (const float*)d_in[5];   // [256,256]
    const float* b2    = (const float*)d_in[6];   // [256]
    const float* Wc    = (const float*)d_in[7];   // [256,64]
    const float* bc    = (const float*)d_in[8];   // [64]
    float*       out   = (float*)d_out;

    const int N = in_sizes[0] / 128;              // 100000
    const int E = in_sizes[1] / 2;                // 1600000
    const int G = out_size / 64;                  // 2048
    const int K1 = 128, H = 256;

    const int* src = edges;
    const int* dst = edges + E;

    // workspace carve-up
    char* w = (char*)d_ws;
    size_t off = 0;
    float* dinv = (float*)(w + off); off += ((size_t)N * 4 + 255) & ~(size_t)255;
    float* bufA = (float*)(w + off); off += (size_t)N * H * 4;   // h (GEMM out)
    float* bufB = (float*)(w + off); off += (size_t)N * H * 4;   // agg
    float* sums = (float*)(w + off); off += (size_t)G * H * 4;
    float* cnt  = (float*)(w + off); off += ((size_t)G * 4 + 255) & ~(size_t)255;
    float* gbuf = (float*)(w + off); off += (size_t)G * H * 4;
    (void)ws_size; (void)n_in;

    const int T = 256;
    const int totNH = N * H;                      // 25.6M (< INT_MAX)
    const int totGH = G * H;

    // 1) symmetric-norm degrees
    gcn_init_deg   <<<(N + T - 1) / T, T, 0, stream>>>(dinv, N);
    gcn_count_deg  <<<(E + T - 1) / T, T, 0, stream>>>(dst, dinv, E);
    gcn_finish_dinv<<<(N + T - 1) / T, T, 0, stream>>>(dinv, N);

    const int gemmGrid = (N + 31) / 32;
    const int edgeGrid = (int)(((size_t)E * 32 + T - 1) / T);
    const int nodeGrid = (int)(((size_t)N * 32 + T - 1) / T);

    // 2) layer 1: h1 = x @ W1 ; agg1 = D^-1/2 A D^-1/2 h1 + b1
    gcn_gemm_wmma_f32<<<gemmGrid, T, 0, stream>>>(x, W1, bufA, N, K1, /*relu*/0);
    gcn_init_agg     <<<(totNH + T - 1) / T, T, 0, stream>>>(bufA, dinv, b1, bufB, totNH);
    gcn_scatter_edges<<<edgeGrid, T, 0, stream>>>(bufA, bufB, src, dst, dinv, E);

    // 3) layer 2: h2 = relu(agg1) @ W2 ; agg2 = norm-aggregate + b2
    gcn_gemm_wmma_f32<<<gemmGrid, T, 0, stream>>>(bufB, W2, bufA, N, H, /*relu*/1);
    gcn_init_agg     <<<(totNH + T - 1) / T, T, 0, stream>>>(bufA, dinv, b2, bufB, totNH);
    gcn_scatter_edges<<<edgeGrid, T, 0, stream>>>(bufA, bufB, src, dst, dinv, E);

    // 4) mean pool over relu(agg2)
    gcn_zero    <<<(totGH + T - 1) / T, T, 0, stream>>>(sums, totGH);
    gcn_zero    <<<(G + T - 1) / T, T, 0, stream>>>(cnt, G);
    gcn_pool    <<<nodeGrid, T, 0, stream>>>(bufB, batch, sums, cnt, N);
    gcn_pool_div<<<(totGH + T - 1) / T, T, 0, stream>>>(sums, cnt, gbuf, totGH);

    // 5) classifier + log_softmax
    gcn_classifier<<<(int)(((size_t)G * 32 + T - 1) / T), T, 0, stream>>>(gbuf, Wc, bc, out, G);
}